// GNModel_9844065042805
// MI455X (gfx1250) — compile-verified
//
#include <hip/hip_runtime.h>
#include <hip/hip_bf16.h>

#define HID 6
#define NUM_G 1000

typedef float v2f __attribute__((ext_vector_type(2)));
typedef float v8f __attribute__((ext_vector_type(8)));

// ---------------------------------------------------------------------------
// Edge phase: m = relu(h[src] + e);  agg[dst] += m   (6 f32 atomics per edge)
// edge_attr / edge_index are streamed once per layer and exceed the 192MB L2,
// so load them non-temporally; h gather + agg atomics stay RT (L2-resident,
// h + agg = 48MB << 192MB L2). All indexing is 32-bit (max index 72M).
// ---------------------------------------------------------------------------
__global__ void edge_scatter(const float* __restrict__ h,
                             const float* __restrict__ eattr,
                             const int*   __restrict__ ei,
                             float*       __restrict__ agg,
                             int E) {
    int t = blockIdx.x * blockDim.x + threadIdx.x;
    if (t >= E) return;
    int s = __builtin_nontemporal_load(ei + t);
    int d = __builtin_nontemporal_load(ei + E + t);

    // rows are 24B and 8B-aligned -> three b64 loads each
    const v2f* hr = (const v2f*)(h + s * HID);
    const v2f* er = (const v2f*)(eattr + t * HID);
    float*     ar = agg + d * HID;

    v2f h0 = hr[0], h1 = hr[1], h2 = hr[2];
    v2f e0 = __builtin_nontemporal_load(er + 0);
    v2f e1 = __builtin_nontemporal_load(er + 1);
    v2f e2 = __builtin_nontemporal_load(er + 2);

    float m0 = fmaxf(h0.x + e0.x, 0.f);
    float m1 = fmaxf(h0.y + e0.y, 0.f);
    float m2 = fmaxf(h1.x + e1.x, 0.f);
    float m3 = fmaxf(h1.y + e1.y, 0.f);
    float m4 = fmaxf(h2.x + e2.x, 0.f);
    float m5 = fmaxf(h2.y + e2.y, 0.f);

    atomicAdd(ar + 0, m0);
    atomicAdd(ar + 1, m1);
    atomicAdd(ar + 2, m2);
    atomicAdd(ar + 3, m3);
    atomicAdd(ar + 4, m4);
    atomicAdd(ar + 5, m5);
}

// ---------------------------------------------------------------------------
// Node phase: out = relu?((h + agg) @ W + b) for a 16-node tile per wave32,
// via two chained exact-fp32 V_WMMA_F32_16X16X4_F32 (K = 6 padded to 8).
//
// A (16x4 f32): lanes 0-15 hold rows M=0..15, VGPR0=K=kb, VGPR1=K=kb+1,
//               where kb = 0 for lanes 0-15 and kb = 2 for lanes 16-31.
// B (4x16 f32): VGPR0: lanes 0-15 -> K=kb,N=lane; symmetric to A.
// C/D (16x16 f32, 8 VGPRs): VGPR r: lanes 0-15 -> M=r,N=lane;
//                            lanes 16-31 -> M=r+8,N=lane-16.
// ---------------------------------------------------------------------------
__global__ void node_update_wmma(const float* __restrict__ h,
                                 const float* __restrict__ agg,
                                 const float* __restrict__ W,    // 6x6 row-major
                                 const float* __restrict__ bias, // 6
                                 float*       __restrict__ out,
                                 int N, int do_relu) {
    const int lane = threadIdx.x & 31;
    const int wave = threadIdx.x >> 5;
    const int wavesPerBlock = blockDim.x >> 5;
    const int tile  = blockIdx.x * wavesPerBlock + wave;
    const int node0 = tile * 16;

    const int m  = lane & 15;   // A: row in tile; B/D: column N
    const int hi = lane >> 4;   // lane half selects K pair / row half
    const int kb = hi * 2;

    const int  node   = node0 + m;
    const bool nodeOK = (node < N);

    // B operands (weights, zero-padded 6x6 -> 4x16 tiles for K-steps 0 and 4)
    v2f b0, b1;
    b0.x = (m < HID) ? W[(kb + 0) * HID + m] : 0.f;          // k = kb   (0..3 < 6)
    b0.y = (m < HID) ? W[(kb + 1) * HID + m] : 0.f;          // k = kb+1
    b1.x = (m < HID && 4 + kb + 0 < HID) ? W[(4 + kb + 0) * HID + m] : 0.f;
    b1.y = (m < HID && 4 + kb + 1 < HID) ? W[(4 + kb + 1) * HID + m] : 0.f;

    // A operands: (h + agg)[node][k], zero-padded past K=5 / past N nodes.
    // Row base = node*24B is 8B aligned; kb offset is 0 or 8B -> b64 loads.
    v2f a0 = {0.f, 0.f}, a1 = {0.f, 0.f};
    if (nodeOK) {
        const int base = node * HID;
        v2f x0 = *(const v2f*)(h + base + kb);
        v2f y0 = *(const v2f*)(agg + base + kb);
        a0.x = x0.x + y0.x;
        a0.y = x0.y + y0.y;
        if (hi == 0) {                       // k = 4,5 valid; k = 6,7 stay zero
            v2f x1 = *(const v2f*)(h + base + 4);
            v2f y1 = *(const v2f*)(agg + base + 4);
            a1.x = x1.x + y1.x;
            a1.y = x1.y + y1.y;
        }
    }

    v8f c = {};
    // 8 args: (neg_a, A, neg_b, B, c_mod, C, reuse_a, reuse_b)
    c = __builtin_amdgcn_wmma_f32_16x16x4_f32(false, a0, false, b0, (short)0, c, false, false);
    c = __builtin_amdgcn_wmma_f32_16x16x4_f32(false, a1, false, b1, (short)0, c, false, false);

    // Write back: this lane holds column N=m for rows r + hi*8 of the tile.
    if (m < HID) {
        const float bn   = bias[m];
        const int   row0 = node0 + hi * 8;
        float* op = out + row0 * HID + m;
        if (node0 + 16 <= N) {               // full tile: unconditional stores
#pragma unroll
            for (int r = 0; r < 8; ++r) {
                float v = c[r] + bn;
                if (do_relu) v = fmaxf(v, 0.f);
                op[r * HID] = v;
            }
        } else {                             // ragged tail tile
#pragma unroll
            for (int r = 0; r < 8; ++r) {
                if (row0 + r < N) {
                    float v = c[r] + bn;
                    if (do_relu) v = fmaxf(v, 0.f);
                    op[r * HID] = v;
                }
            }
        }
    }
}

// ---------------------------------------------------------------------------
// Global mean pool: per-node scatter-add into 1000 graph slots (L2-resident).
// ---------------------------------------------------------------------------
__global__ void pool_scatter(const float* __restrict__ h,
                             const int*   __restrict__ batch,
                             float* __restrict__ sums,
                             float* __restrict__ cnts, int N) {
    int t = blockIdx.x * blockDim.x + threadIdx.x;
    if (t >= N) return;
    int g = __builtin_nontemporal_load(batch + t);
    const v2f* hr = (const v2f*)(h + t * HID);
    v2f h0 = hr[0], h1 = hr[1], h2 = hr[2];
    float* sg = sums + g * HID;
    atomicAdd(sg + 0, h0.x);
    atomicAdd(sg + 1, h0.y);
    atomicAdd(sg + 2, h1.x);
    atomicAdd(sg + 3, h1.y);
    atomicAdd(sg + 4, h2.x);
    atomicAdd(sg + 5, h2.y);
    atomicAdd(&cnts[g], 1.0f);
}

__global__ void final_linear(const float* __restrict__ sums,
                             const float* __restrict__ cnts,
                             const float* __restrict__ Wl,
                             const float* __restrict__ bl,
                             float* __restrict__ out, int G) {
    int g = blockIdx.x * blockDim.x + threadIdx.x;
    if (g >= G) return;
    float inv = 1.0f / fmaxf(cnts[g], 1.0f);
    float acc = bl[0];
#pragma unroll
    for (int j = 0; j < HID; ++j) acc += sums[g * HID + j] * inv * Wl[j];
    out[g] = acc;
}

// ---------------------------------------------------------------------------
extern "C" void kernel_launch(void* const* d_in, const int* in_sizes, int n_in,
                              void* d_out, int out_size, void* d_ws, size_t ws_size,
                              hipStream_t stream) {
    const float* x     = (const float*)d_in[0];
    const int*   ei    = (const int*)  d_in[1];
    const float* eattr = (const float*)d_in[2];
    const int*   batch = (const int*)  d_in[3];
    const float* W1 = (const float*)d_in[4];  const float* b1 = (const float*)d_in[5];
    const float* W2 = (const float*)d_in[6];  const float* b2 = (const float*)d_in[7];
    const float* W3 = (const float*)d_in[8];  const float* b3 = (const float*)d_in[9];
    const float* Wl = (const float*)d_in[10]; const float* bl = (const float*)d_in[11];
    float* out = (float*)d_out;

    const int N = in_sizes[0] / HID;
    const int E = in_sizes[2] / HID;
    const int G = out_size;                 // 1000 graphs, output [G,1]

    float* ws   = (float*)d_ws;
    float* agg  = ws;
    float* hA   = agg + (size_t)N * HID;
    float* hB   = hA  + (size_t)N * HID;
    float* sums = hB  + (size_t)N * HID;
    float* cnts = sums + (size_t)G * HID;

    const int EB = 256;
    const dim3 eg((E + EB - 1) / EB), eb(EB);
    const int nodesPerBlock = (256 / 32) * 16;   // 8 waves x 16 nodes
    const dim3 ng((N + nodesPerBlock - 1) / nodesPerBlock), nb(256);

    auto layer = [&](const float* hin, const float* W, const float* b,
                     float* hout, int relu) {
        hipMemsetAsync(agg, 0, (size_t)N * HID * sizeof(float), stream);
        edge_scatter<<<eg, eb, 0, stream>>>(hin, eattr, ei, agg, E);
        node_update_wmma<<<ng, nb, 0, stream>>>(hin, agg, W, b, hout, N, relu);
    };

    layer(x,  W1, b1, hA, 1);
    layer(hA, W2, b2, hB, 1);
    layer(hB, W3, b3, hA, 0);

    hipMemsetAsync(sums, 0, (size_t)(G * HID + G) * sizeof(float), stream);
    pool_scatter<<<(N + 255) / 256, 256, 0, stream>>>(hA, batch, sums, cnts, N);
    final_linear<<<(G + 255) / 256, 256, 0, stream>>>(sums, cnts, Wl, bl, out, G);
}